// EMOGConv_14963666059792
// MI455X (gfx1250) — compile-verified
//
#include <hip/hip_runtime.h>
#include <math.h>

// ---------------- constants ----------------
#define BB 8
#define NN 1024
#define KK 20
#define EPSf    1e-6f
#define BN_EPSf 1e-5f
#define SLOPEf  0.2f

typedef float v2f __attribute__((ext_vector_type(2)));
typedef float v8f __attribute__((ext_vector_type(8)));

// ---------------- static device workspace (~135MB) ----------------
// gram 8M, xx 8K, idx 160K, xc 4.15M, x5 8.38M, xm 8K, z 8.38M, z2 4.18M,
// z0t 73K, g 32K, h1/h2, stats  => 33,774,892 floats total
__device__ float g_buf[33800000];

// =================================================================
// Gram matrix  G[b][i][j] = sum_f feat[f][i]*feat[f][j]  via WMMA f32 16x16x4
// feat[f][n] = xin[b*bstride + (choff*3 + f)*NN + n],   f in [0, d)
// One wave per 16x16 tile.  Main K-loop is branch-free (uniform trip count);
// the ragged tail (d % 4) is handled once with predicated loads so the hot
// loop never touches EXEC.
// =================================================================
__global__ void __launch_bounds__(32)
gram_kernel(const float* __restrict__ xin, int bstride, int choff,
            int d, float* __restrict__ gram) {
  const int b  = blockIdx.z;
  const int i0 = blockIdx.x << 4;
  const int j0 = blockIdx.y << 4;
  const int lane = threadIdx.x & 31;
  const int half = lane >> 4;       // 0: lanes 0-15, 1: lanes 16-31
  const int lr   = lane & 15;
  const float* base = xin + (size_t)b * bstride + (size_t)choff * 3 * NN;
  // lane-resident base pointers: row (2*half) at column i0+lr / j0+lr
  const float* pa = base + (size_t)(2 * half) * NN + i0 + lr;
  const float* pb = base + (size_t)(2 * half) * NN + j0 + lr;

  v8f acc;
#pragma unroll
  for (int r = 0; r < 8; ++r) acc[r] = 0.0f;

  int kk = 0;
  for (; kk + 4 <= d; kk += 4) {           // branch-free main loop
    v2f a, bm;
    a.x  = pa[(size_t)kk * NN];
    a.y  = pa[(size_t)kk * NN + NN];
    bm.x = pb[(size_t)kk * NN];
    bm.y = pb[(size_t)kk * NN + NN];
    acc = __builtin_amdgcn_wmma_f32_16x16x4_f32(false, a, false, bm,
                                                (short)0, acc, false, false);
  }
  if (kk < d) {                             // predicated tail (once)
    const int ka = kk + 2 * half;
    v2f a, bm;
    a.x  = (ka     < d) ? pa[(size_t)kk * NN]      : 0.0f;
    a.y  = (ka + 1 < d) ? pa[(size_t)kk * NN + NN] : 0.0f;
    bm.x = (ka     < d) ? pb[(size_t)kk * NN]      : 0.0f;
    bm.y = (ka + 1 < d) ? pb[(size_t)kk * NN + NN] : 0.0f;
    acc = __builtin_amdgcn_wmma_f32_16x16x4_f32(false, a, false, bm,
                                                (short)0, acc, false, false);
  }
  // D layout: VGPR r -> M=r (lanes 0-15) / M=r+8 (lanes 16-31), N=lane%16
  float* go = gram + (size_t)b * NN * NN;
#pragma unroll
  for (int r = 0; r < 8; ++r) {
    const int m = half * 8 + r;
    go[(size_t)(i0 + m) * NN + j0 + lr] = acc[r];
  }
}

// ---------------- per-point squared norms ----------------
__global__ void xx_kernel(const float* __restrict__ xin, int bstride, int choff,
                          int d, float* __restrict__ xx) {
  int t = blockIdx.x * blockDim.x + threadIdx.x;
  if (t >= BB * NN) return;
  int b = t / NN, n = t % NN;
  const float* base = xin + (size_t)b * bstride + (size_t)choff * 3 * NN;
  float s = 0.0f;
  for (int f = 0; f < d; ++f) { float v = base[(size_t)f * NN + n]; s += v * v; }
  xx[t] = s;
}

// ---------------- tag-masked top-K (K=20) per row ----------------
__global__ void topk_kernel(const float* __restrict__ gram, const float* __restrict__ xx,
                            const int* __restrict__ tag, int* __restrict__ idx) {
  const int i = blockIdx.x, b = blockIdx.y;
  __shared__ float nd[NN];
  __shared__ float bv[128];
  __shared__ int   bi[128];
  const int tid = threadIdx.x;
  const float* gr = gram + ((size_t)b * NN + i) * NN;
  const float xi = xx[b * NN + i];
  const int   ti = tag[b * NN + i];
  for (int j = tid; j < NN; j += 128) {
    float v = 2.0f * gr[j] - xi - xx[b * NN + j];
    nd[j] = (tag[b * NN + j] == ti) ? v : -INFINITY;
  }
  __syncthreads();
  for (int k = 0; k < KK; ++k) {
    float best = -INFINITY; int bj = NN;
    for (int j = tid; j < NN; j += 128) {
      float v = nd[j];
      if (v > best || (v == best && j < bj)) { best = v; bj = j; }
    }
    bv[tid] = best; bi[tid] = bj;
    __syncthreads();
    for (int s = 64; s > 0; s >>= 1) {
      if (tid < s) {
        if (bv[tid + s] > bv[tid] ||
            (bv[tid + s] == bv[tid] && bi[tid + s] < bi[tid])) {
          bv[tid] = bv[tid + s]; bi[tid] = bi[tid + s];
        }
      }
      __syncthreads();
    }
    if (tid == 0) { idx[((size_t)b * NN + i) * KK + k] = bi[0]; nd[bi[0]] = -INFINITY; }
    __syncthreads();
  }
}

// ---------------- small helpers ----------------
__global__ void zero_kernel(float* p, int n) {
  int t = blockIdx.x * blockDim.x + threadIdx.x;
  if (t < n) p[t] = 0.0f;
}

__global__ void bn_finalize(const float* __restrict__ s1, const float* __restrict__ s2,
                            int O, float inv_cnt, float* __restrict__ mu,
                            float* __restrict__ istd) {
  int o = blockIdx.x * blockDim.x + threadIdx.x;
  if (o >= O) return;
  float m = s1[o] * inv_cnt;
  float v = s2[o] * inv_cnt - m * m;
  mu[o] = m;
  istd[o] = rsqrtf(v + BN_EPSf);
}

// =================================================================
// Edge-conv stage, pass A: accumulate per-channel ||p|| statistics.
// e[i<C] = nb - ctr,  e[C+i] = ctr  (channels), 3 comps each.
// =================================================================
__global__ void edge_statsA(const float* __restrict__ xin, int bstride, int choff,
                            int C, int O, const float* __restrict__ wf,
                            const int* __restrict__ idx,
                            float* __restrict__ s1, float* __restrict__ s2) {
  const int n = blockIdx.x, b = blockIdx.y;
  const int F = 3 * C;
  __shared__ float ctr[128];
  __shared__ float nbr[KK][128];
  __shared__ float a1[85], a2[85];
  const int tid = threadIdx.x;
  const float* base = xin + (size_t)b * bstride + (size_t)choff * 3 * NN;
  for (int f = tid; f < F; f += blockDim.x) ctr[f] = base[(size_t)f * NN + n];
  const int* id = idx + ((size_t)b * NN + n) * KK;
  for (int t = tid; t < KK * F; t += blockDim.x) {
    int k = t / F, f = t % F;
    nbr[k][f] = base[(size_t)f * NN + id[k]];
  }
  for (int o = tid; o < O; o += blockDim.x) { a1[o] = 0.0f; a2[o] = 0.0f; }
  __syncthreads();
  for (int t = tid; t < O * KK; t += blockDim.x) {
    int o = t / KK, k = t % KK;
    const float* w = wf + (size_t)o * (2 * C);
    float px = 0, py = 0, pz = 0;
    for (int i = 0; i < C; ++i) {
      float c0 = ctr[i*3+0], c1 = ctr[i*3+1], c2 = ctr[i*3+2];
      float e0 = nbr[k][i*3+0] - c0, e1 = nbr[k][i*3+1] - c1, e2 = nbr[k][i*3+2] - c2;
      float wi = w[i], wc = w[C + i];
      px += wi * e0 + wc * c0; py += wi * e1 + wc * c1; pz += wi * e2 + wc * c2;
    }
    float nr = sqrtf(px*px + py*py + pz*pz) + EPSf;
    atomicAdd(&a1[o], nr);
    atomicAdd(&a2[o], nr * nr);
  }
  __syncthreads();
  for (int o = tid; o < O; o += blockDim.x) {
    atomicAdd(&s1[o], a1[o]);
    atomicAdd(&s2[o], a2[o]);
  }
}

// =================================================================
// Edge-conv stage, pass B: recompute p,d; BN + VN-LeakyReLU; VN-maxpool over K.
// =================================================================
__global__ void edge_passB(const float* __restrict__ xin, int bstride, int choff,
                           int C, int O, const float* __restrict__ wf,
                           const float* __restrict__ wd,
                           const float* __restrict__ ga, const float* __restrict__ be,
                           const float* __restrict__ mu, const float* __restrict__ istd,
                           const float* __restrict__ wp, const int* __restrict__ idx,
                           float* __restrict__ xout, int obstride, int ochoff) {
  const int n = blockIdx.x, b = blockIdx.y;
  const int F = 3 * C;
  __shared__ float ctr[128];
  __shared__ float nbr[KK][128];
  __shared__ float hsh[85][KK][3];
  __shared__ float dotp[85][KK];
  const int tid = threadIdx.x;
  const float* base = xin + (size_t)b * bstride + (size_t)choff * 3 * NN;
  for (int f = tid; f < F; f += blockDim.x) ctr[f] = base[(size_t)f * NN + n];
  const int* id = idx + ((size_t)b * NN + n) * KK;
  for (int t = tid; t < KK * F; t += blockDim.x) {
    int k = t / F, f = t % F;
    nbr[k][f] = base[(size_t)f * NN + id[k]];
  }
  __syncthreads();
  for (int t = tid; t < O * KK; t += blockDim.x) {
    int o = t / KK, k = t % KK;
    const float* w = wf + (size_t)o * (2 * C);
    const float* v = wd + (size_t)o * (2 * C);
    float px=0,py=0,pz=0, dx=0,dy=0,dz=0;
    for (int i = 0; i < C; ++i) {
      float c0 = ctr[i*3+0], c1 = ctr[i*3+1], c2 = ctr[i*3+2];
      float e0 = nbr[k][i*3+0] - c0, e1 = nbr[k][i*3+1] - c1, e2 = nbr[k][i*3+2] - c2;
      float wi = w[i], wc = w[C + i], vi = v[i], vc = v[C + i];
      px += wi*e0 + wc*c0; py += wi*e1 + wc*c1; pz += wi*e2 + wc*c2;
      dx += vi*e0 + vc*c0; dy += vi*e1 + vc*c1; dz += vi*e2 + vc*c2;
    }
    float nr  = sqrtf(px*px + py*py + pz*pz) + EPSf;
    float fac = ((nr - mu[o]) * istd[o] * ga[o] + be[o]) / nr;
    px *= fac; py *= fac; pz *= fac;
    float dot = px*dx + py*dy + pz*dz;
    float dsq = dx*dx + dy*dy + dz*dz;
    float qx = px, qy = py, qz = pz;
    if (dot < 0.0f) { float s = dot / (dsq + EPSf); qx -= s*dx; qy -= s*dy; qz -= s*dz; }
    hsh[o][k][0] = SLOPEf*px + (1.0f - SLOPEf)*qx;
    hsh[o][k][1] = SLOPEf*py + (1.0f - SLOPEf)*qy;
    hsh[o][k][2] = SLOPEf*pz + (1.0f - SLOPEf)*qz;
  }
  __syncthreads();
  for (int t = tid; t < O * KK; t += blockDim.x) {
    int o = t / KK, k = t % KK;
    const float* wr = wp + (size_t)o * O;
    float d0=0,d1=0,d2=0;
    for (int q = 0; q < O; ++q) {
      float wv = wr[q];
      d0 += wv * hsh[q][k][0]; d1 += wv * hsh[q][k][1]; d2 += wv * hsh[q][k][2];
    }
    dotp[o][k] = hsh[o][k][0]*d0 + hsh[o][k][1]*d1 + hsh[o][k][2]*d2;
  }
  __syncthreads();
  for (int o = tid; o < O; o += blockDim.x) {
    int kb = 0; float bestv = dotp[o][0];
    for (int k = 1; k < KK; ++k)
      if (dotp[o][k] > bestv) { bestv = dotp[o][k]; kb = k; }
    float* op = xout + (size_t)b * obstride + (size_t)(ochoff + o) * 3 * NN + n;
    op[0]      = hsh[o][kb][0];
    op[NN]     = hsh[o][kb][1];
    op[2 * NN] = hsh[o][kb][2];
  }
}

// =================================================================
// VN linear layer (no K, no pool): c5 / s1 / s2.
// Input column i: i<I1 -> in1[b,i,:,n], else -> in2mean[b,i-I1,:]
// =================================================================
__global__ void vn_statsA(const float* __restrict__ in1, int b1stride, int I1,
                          const float* __restrict__ in2mean, int Itot, int O,
                          const float* __restrict__ wf,
                          float* __restrict__ s1, float* __restrict__ s2) {
  const int n = blockIdx.x, b = blockIdx.y;
  __shared__ float col[2046];
  const int tid = threadIdx.x;
  const int I2 = Itot - I1;
  for (int t = tid; t < Itot * 3; t += blockDim.x) {
    int i = t / 3, cmp = t % 3;
    col[t] = (i < I1)
      ? in1[(size_t)b * b1stride + (size_t)i * 3 * NN + cmp * NN + n]
      : in2mean[((size_t)b * I2 + (i - I1)) * 3 + cmp];
  }
  __syncthreads();
  for (int o = tid; o < O; o += blockDim.x) {
    const float* w = wf + (size_t)o * Itot;
    float px=0,py=0,pz=0;
    for (int i = 0; i < Itot; ++i) {
      float wv = w[i];
      px += wv*col[i*3+0]; py += wv*col[i*3+1]; pz += wv*col[i*3+2];
    }
    float nr = sqrtf(px*px + py*py + pz*pz) + EPSf;
    atomicAdd(&s1[o], nr);
    atomicAdd(&s2[o], nr * nr);
  }
}

__global__ void vn_passB(const float* __restrict__ in1, int b1stride, int I1,
                         const float* __restrict__ in2mean, int Itot, int O,
                         const float* __restrict__ wf, const float* __restrict__ wd,
                         int Od, const float* __restrict__ ga, const float* __restrict__ be,
                         const float* __restrict__ mu, const float* __restrict__ istd,
                         float* __restrict__ out, int obstride) {
  const int n = blockIdx.x, b = blockIdx.y;
  __shared__ float col[2046];
  const int tid = threadIdx.x;
  const int I2 = Itot - I1;
  for (int t = tid; t < Itot * 3; t += blockDim.x) {
    int i = t / 3, cmp = t % 3;
    col[t] = (i < I1)
      ? in1[(size_t)b * b1stride + (size_t)i * 3 * NN + cmp * NN + n]
      : in2mean[((size_t)b * I2 + (i - I1)) * 3 + cmp];
  }
  __syncthreads();
  for (int o = tid; o < O; o += blockDim.x) {
    const float* w = wf + (size_t)o * Itot;
    const float* v = wd + (size_t)(Od == 1 ? 0 : o) * Itot;
    float px=0,py=0,pz=0,dx=0,dy=0,dz=0;
    for (int i = 0; i < Itot; ++i) {
      float c0 = col[i*3+0], c1 = col[i*3+1], c2 = col[i*3+2];
      float wv = w[i], vv = v[i];
      px += wv*c0; py += wv*c1; pz += wv*c2;
      dx += vv*c0; dy += vv*c1; dz += vv*c2;
    }
    float nr  = sqrtf(px*px + py*py + pz*pz) + EPSf;
    float fac = ((nr - mu[o]) * istd[o] * ga[o] + be[o]) / nr;
    px *= fac; py *= fac; pz *= fac;
    float dot = px*dx + py*dy + pz*dz;
    float dsq = dx*dx + dy*dy + dz*dz;
    float qx = px, qy = py, qz = pz;
    if (dot < 0.0f) { float s = dot / (dsq + EPSf); qx -= s*dx; qy -= s*dy; qz -= s*dz; }
    float* op = out + (size_t)b * obstride + (size_t)o * 3 * NN + n;
    op[0]      = SLOPEf*px + (1.0f - SLOPEf)*qx;
    op[NN]     = SLOPEf*py + (1.0f - SLOPEf)*qy;
    op[2 * NN] = SLOPEf*pz + (1.0f - SLOPEf)*qz;
  }
}

// ---------------- mean over N of x5 -> xm[b][o][c] ----------------
__global__ void mean_kernel(const float* __restrict__ x5, float* __restrict__ xm) {
  int t = blockIdx.x * blockDim.x + threadIdx.x;
  if (t >= BB * 341 * 3) return;
  int b = t / (341 * 3), r = t % (341 * 3), o = r / 3, cmp = r % 3;
  const float* p = x5 + (size_t)b * 341 * 3 * NN + (size_t)o * 3 * NN + cmp * NN;
  float s = 0.0f;
  for (int n = 0; n < NN; ++n) s += p[n];
  xm[t] = s / (float)NN;
}

// ---------------- z0t[b][j][k][n] = sum_i slin[k,i]*z2[b,i,j,n] ----------------
__global__ void z0_kernel(const float* __restrict__ z2, const float* __restrict__ slin,
                          float* __restrict__ z0t) {
  int t = blockIdx.x * blockDim.x + threadIdx.x;
  if (t >= BB * 9 * NN) return;
  int n = t % NN; int r = t / NN;
  int k = r % 3; r /= 3;
  int j = r % 3; int b = r / 3;
  float s = 0.0f;
  for (int i = 0; i < 170; ++i)
    s += slin[k * 170 + i] * z2[((size_t)b * 170 + i) * 3 * NN + j * NN + n];
  z0t[t] = s;
}

// ---------------- xs = xcat x z0t, reduce max+mean over N -> g[b][4092] ----------------
__global__ void g_kernel(const float* __restrict__ x5, const float* __restrict__ xm,
                         const float* __restrict__ z0t, float* __restrict__ g) {
  int t = blockIdx.x * blockDim.x + threadIdx.x;
  if (t >= BB * 682 * 3) return;
  int b = t / (682 * 3), r = t % (682 * 3), i = r / 3, kc = r % 3;
  float mx = -INFINITY, sm = 0.0f;
  for (int n = 0; n < NN; ++n) {
    float v = 0.0f;
    for (int j = 0; j < 3; ++j) {
      float xc = (i < 341)
        ? x5[((size_t)b * 341 + i) * 3 * NN + j * NN + n]
        : xm[((size_t)b * 341 + (i - 341)) * 3 + j];
      v += xc * z0t[(((size_t)b * 3 + j) * 3 + kc) * NN + n];
    }
    mx = fmaxf(mx, v); sm += v;
  }
  g[b * 4092 + i * 3 + kc]        = mx;
  g[b * 4092 + 2046 + i * 3 + kc] = sm / (float)NN;
}

// ---------------- tiny MLP layer (+ optional bn over batch + leaky) ----------------
__global__ void mlp_kernel(const float* __restrict__ in, int inF,
                           const float* __restrict__ w, const float* __restrict__ bias,
                           const float* __restrict__ bg, const float* __restrict__ bb,
                           int outF, float* __restrict__ out, int doBnRelu) {
  int f = blockIdx.x * blockDim.x + threadIdx.x;
  if (f >= outF) return;
  float h[BB];
  for (int b = 0; b < BB; ++b) {
    float s = bias[f];
    const float* wr = w + (size_t)f * inF;
    const float* ir = in + (size_t)b * inF;
    for (int i = 0; i < inF; ++i) s += wr[i] * ir[i];
    h[b] = s;
  }
  if (doBnRelu) {
    float mu = 0.0f;
    for (int b = 0; b < BB; ++b) mu += h[b];
    mu *= (1.0f / BB);
    float var = 0.0f;
    for (int b = 0; b < BB; ++b) { float d = h[b] - mu; var += d * d; }
    var *= (1.0f / BB);
    float is = rsqrtf(var + BN_EPSf);
    for (int b = 0; b < BB; ++b) {
      float v = (h[b] - mu) * is * bg[f] + bb[f];
      h[b] = (v >= 0.0f) ? v : SLOPEf * v;
    }
  }
  for (int b = 0; b < BB; ++b) out[b * outF + f] = h[b];
}

// =================================================================
// host orchestration
// =================================================================
extern "C" void kernel_launch(void* const* d_in, const int* in_sizes, int n_in,
                              void* d_out, int out_size, void* d_ws, size_t ws_size,
                              hipStream_t stream) {
  (void)in_sizes; (void)n_in; (void)out_size; (void)d_ws; (void)ws_size;

  float* W = nullptr;
  hipGetSymbolAddress((void**)&W, HIP_SYMBOL(g_buf));

  size_t off = 0;
  auto take = [&](size_t nf) { float* p = W + off; off += nf; return p; };
  float* gram = take((size_t)BB * NN * NN);
  float* xx   = take((size_t)BB * NN);
  int*   idx  = (int*)take((size_t)BB * NN * KK);
  float* xc   = take((size_t)BB * 169 * 3 * NN);
  float* x5   = take((size_t)BB * 341 * 3 * NN);
  float* xm   = take((size_t)BB * 341 * 3);
  float* z    = take((size_t)BB * 341 * 3 * NN);
  float* z2   = take((size_t)BB * 170 * 3 * NN);
  float* z0t  = take((size_t)BB * 9 * NN);
  float* gb   = take((size_t)BB * 4092);
  float* h1   = take((size_t)BB * 512);
  float* h2   = take((size_t)BB * 256);
  float* st   = take((size_t)4 * 341);
  float* s1v = st, *s2v = st + 341, *muv = st + 682, *isv = st + 1023;

  const float* x   = (const float*)d_in[0];
  const int*   tag = (const int*)d_in[1];
  auto F = [&](int i) { return (const float*)d_in[i]; };

  const int XCB = 169 * 3 * NN;  // xc batch stride
  const dim3 gGram(NN / 16, NN / 16, BB);
  const dim3 gBN(NN, BB);

  // ---- edge-conv stages: (xin,bstride,choff,C,O, wf-idx, wp-idx, out choff)
  struct Stage { const float* xin; int bstr, choff, C, O, wi, pi, oo; };
  Stage stages[4] = {
    { x,  3 * NN, 0,  1,  21, 2,  30, 0  },
    { xc, XCB,    0,  21, 21, 6,  31, 21 },
    { xc, XCB,    21, 21, 42, 10, 32, 42 },
    { xc, XCB,    42, 42, 85, 14, 33, 84 },
  };
  for (int s = 0; s < 4; ++s) {
    const Stage& c = stages[s];
    const int d = 3 * c.C;
    gram_kernel<<<gGram, 32, 0, stream>>>(c.xin, c.bstr, c.choff, d, gram);
    xx_kernel<<<(BB * NN + 255) / 256, 256, 0, stream>>>(c.xin, c.bstr, c.choff, d, xx);
    topk_kernel<<<gBN, 128, 0, stream>>>(gram, xx, tag, idx);
    zero_kernel<<<4, 256, 0, stream>>>(s1v, 682);
    edge_statsA<<<gBN, 256, 0, stream>>>(c.xin, c.bstr, c.choff, c.C, c.O, F(c.wi), idx, s1v, s2v);
    bn_finalize<<<2, 256, 0, stream>>>(s1v, s2v, c.O, 1.0f / (float)(BB * NN * KK), muv, isv);
    edge_passB<<<gBN, 256, 0, stream>>>(c.xin, c.bstr, c.choff, c.C, c.O, F(c.wi), F(c.wi + 1),
                                        F(c.wi + 2), F(c.wi + 3), muv, isv, F(c.pi), idx,
                                        xc, XCB, c.oo);
  }

  // ---- c5: xc(169) -> x5(341), wd broadcast (Od=1)
  zero_kernel<<<4, 256, 0, stream>>>(s1v, 682);
  vn_statsA<<<gBN, 256, 0, stream>>>(xc, XCB, 169, nullptr, 169, 341, F(26), s1v, s2v);
  bn_finalize<<<2, 256, 0, stream>>>(s1v, s2v, 341, 1.0f / (float)(BB * NN), muv, isv);
  vn_passB<<<gBN, 256, 0, stream>>>(xc, XCB, 169, nullptr, 169, 341, F(26), F(27), 1,
                                    F(28), F(29), muv, isv, x5, 341 * 3 * NN);

  // ---- xm = mean_N(x5)
  mean_kernel<<<(BB * 341 * 3 + 255) / 256, 256, 0, stream>>>(x5, xm);

  // ---- s1: xcat(682 = x5 ++ xm) -> z(341)
  zero_kernel<<<4, 256, 0, stream>>>(s1v, 682);
  vn_statsA<<<gBN, 256, 0, stream>>>(x5, 341 * 3 * NN, 341, xm, 682, 341, F(18), s1v, s2v);
  bn_finalize<<<2, 256, 0, stream>>>(s1v, s2v, 341, 1.0f / (float)(BB * NN), muv, isv);
  vn_passB<<<gBN, 256, 0, stream>>>(x5, 341 * 3 * NN, 341, xm, 682, 341, F(18), F(19), 341,
                                    F(20), F(21), muv, isv, z, 341 * 3 * NN);

  // ---- s2: z(341) -> z2(170)
  zero_kernel<<<4, 256, 0, stream>>>(s1v, 682);
  vn_statsA<<<gBN, 256, 0, stream>>>(z, 341 * 3 * NN, 341, nullptr, 341, 170, F(22), s1v, s2v);
  bn_finalize<<<2, 256, 0, stream>>>(s1v, s2v, 170, 1.0f / (float)(BB * NN), muv, isv);
  vn_passB<<<gBN, 256, 0, stream>>>(z, 341 * 3 * NN, 341, nullptr, 341, 170, F(22), F(23), 170,
                                    F(24), F(25), muv, isv, z2, 170 * 3 * NN);

  // ---- z0t, xs-contraction + global max/mean pooling
  z0_kernel<<<(BB * 9 * NN + 255) / 256, 256, 0, stream>>>(z2, F(34), z0t);
  g_kernel<<<(BB * 682 * 3 + 255) / 256, 256, 0, stream>>>(x5, xm, z0t, gb);

  // ---- final MLP
  mlp_kernel<<<2, 256, 0, stream>>>(gb, 4092, F(35), F(36), F(37), F(38), 512, h1, 1);
  mlp_kernel<<<1, 256, 0, stream>>>(h1, 512, F(39), F(40), F(41), F(42), 256, h2, 1);
  mlp_kernel<<<1, 128, 0, stream>>>(h2, 256, F(43), F(44), F(43), F(44), 128, (float*)d_out, 0);
}